// MultiSupConLoss_23381801959424
// MI455X (gfx1250) — compile-verified
//
#include <hip/hip_runtime.h>
#include <hip/hip_bf16.h>
#include <stdint.h>

// MultiSupConLoss, fused single-pass implementation for gfx1250 (MI455X).
// sims GEMM on the f32 WMMA pipe (v_wmma_f32_16x16x4_f32), jaccard mask via
// packed-bit popcount, online log-sum-exp so the 4096x4096 sims matrix is
// never materialized. All exp/log in f32 to reproduce the reference's
// underflow behavior (row max = diagonal ~3657 -> exp_logits == 0 -> log(eps)).

typedef __attribute__((ext_vector_type(2))) float v2f;
typedef __attribute__((ext_vector_type(8))) float v8f;

#define B_ROWS 2048
#define NROWS  4096
#define DDIM   256
#define TEMP   0.07f
#define INV_T  (1.0f / 0.07f)
#define EPSV   1e-8f
#define CTH    0.3f
#define INV_CT (1.0f / (0.3f + 1e-8f))

// contrast_feature row n (n = v*B + b) -> pointer into features[b][v][:]
__device__ __forceinline__ const float* cf_row(const float* f, int n) {
  return f + ((n & (B_ROWS - 1)) * (2 * DDIM) + (n >> 11) * DDIM);
}

__global__ void pack_labels(const int* __restrict__ labels, uint4* __restrict__ bits) {
  int b = blockIdx.x * blockDim.x + threadIdx.x;
  if (b >= B_ROWS) return;
  const int* row = labels + b * 80;
  unsigned w0 = 0, w1 = 0, w2 = 0, cnt = 0;
  #pragma unroll
  for (int i = 0; i < 32; ++i) { unsigned v = (row[i]      != 0); w0 |= v << i; cnt += v; }
  #pragma unroll
  for (int i = 0; i < 32; ++i) { unsigned v = (row[32 + i] != 0); w1 |= v << i; cnt += v; }
  #pragma unroll
  for (int i = 0; i < 16; ++i) { unsigned v = (row[64 + i] != 0); w2 |= v << i; cnt += v; }
  bits[b] = make_uint4(w0, w1, w2, cnt);
}

__global__ void __launch_bounds__(256) supcon_main(const float* __restrict__ feat,
                                                   const uint4* __restrict__ bits,
                                                   float* __restrict__ partial) {
  // A tile staged kpair-major: element (row, kk) at ldsA[(kk>>1)*32 + row*2 + (kk&1)]
  // -> per-lane ds_load_b64 reads are bank-conflict free.
  __shared__ float ldsA[16 * DDIM];
  __shared__ float red[8][16][5];

  const int i0   = blockIdx.x * 16;     // this WG owns rows i0..i0+15
  const int tid  = threadIdx.x;
  const int wave = tid >> 5;
  const int lane = tid & 31;
  const int lrow = lane & 15;
  const int hi16 = lane >> 4;           // 0: lanes 0-15, 1: lanes 16-31

  // Stage the 16-row A strip (16 KB) into LDS once; reused for all 256 j-tiles.
  for (int idx = tid; idx < 16 * DDIM; idx += 256) {
    int row = idx >> 8;
    int kk  = idx & 255;
    ldsA[(kk >> 1) * 32 + row * 2 + (kk & 1)] = cf_row(feat, i0 + row)[kk];
  }
  __syncthreads();

  // Label bitmasks for the 8 rows this lane's accumulator VGPRs cover.
  const int halo = hi16 * 8;            // C/D layout: lanes>=16 hold rows M+8
  uint4 wi[8];
  #pragma unroll
  for (int r = 0; r < 8; ++r) wi[r] = bits[(i0 + r + halo) & (B_ROWS - 1)];

  // Per-lane running state per row: max, sum-exp, sum(w*s), sum(w), sum(mask)
  float mx[8], Sx[8], Ax[8], Qx[8], Mx[8];
  #pragma unroll
  for (int r = 0; r < 8; ++r) { mx[r] = -INFINITY; Sx[r] = 0.f; Ax[r] = 0.f; Qx[r] = 0.f; Mx[r] = 0.f; }

  const int aBase = hi16 * 32 + lrow * 2;  // float index; + (k>>1)*32 per k-step

  for (int jt = wave; jt < NROWS / 16; jt += 8) {
    const int j0   = jt * 16;
    const int jCol = j0 + lrow;
    const float* pB = cf_row(feat, jCol) + 2 * hi16;

    v8f c = {};
    #pragma unroll 8
    for (int k = 0; k < DDIM; k += 4) {
      v2f a = *(const v2f*)&ldsA[aBase + (k >> 1) * 32];
      v2f b = *(const v2f*)(pB + k);
      // D = A(16x4,f32) * B(4x16,f32) + C ; 64 of these accumulate full K=256
      c = __builtin_amdgcn_wmma_f32_16x16x4_f32(false, a, false, b, (short)0, c, false, false);
    }

    uint4 wj = bits[jCol & (B_ROWS - 1)];
    #pragma unroll
    for (int r = 0; r < 8; ++r) {
      float s    = c[r] * INV_T;                       // sims[iRow][jCol]
      int   iRow = i0 + r + halo;
      int inter  = __popc(wi[r].x & wj.x) + __popc(wi[r].y & wj.y) + __popc(wi[r].z & wj.z);
      float uni  = (float)(int)(wi[r].w + wj.w) - (float)inter;
      float jac  = (float)inter / (uni + EPSV);

      float nm = fmaxf(mx[r], s);                      // max includes diagonal
      float e  = (iRow == jCol) ? 0.0f : __expf(s - nm);
      Sx[r] = Sx[r] * __expf(mx[r] - nm) + e;
      mx[r] = nm;

      if (jac >= CTH && iRow != jCol) {
        float w = jac * INV_CT;
        Ax[r] = fmaf(w, s, Ax[r]);
        Qx[r] += w;
        Mx[r] += 1.0f;
      }
    }
  }

  // Reduce over the 16 lanes of each half (rows r and r+8 are in separate halves).
  #pragma unroll
  for (int r = 0; r < 8; ++r) {
    #pragma unroll
    for (int off = 1; off < 16; off <<= 1) {
      float m2 = __shfl_xor(mx[r], off, 32);
      float S2 = __shfl_xor(Sx[r], off, 32);
      float nm = fmaxf(mx[r], m2);
      Sx[r] = Sx[r] * __expf(mx[r] - nm) + S2 * __expf(m2 - nm);
      mx[r] = nm;
      Ax[r] += __shfl_xor(Ax[r], off, 32);
      Qx[r] += __shfl_xor(Qx[r], off, 32);
      Mx[r] += __shfl_xor(Mx[r], off, 32);
    }
  }

  if (lane == 0 || lane == 16) {
    #pragma unroll
    for (int r = 0; r < 8; ++r) {
      int row = r + halo;
      red[wave][row][0] = mx[r];
      red[wave][row][1] = Sx[r];
      red[wave][row][2] = Ax[r];
      red[wave][row][3] = Qx[r];
      red[wave][row][4] = Mx[r];
    }
  }
  __syncthreads();

  // Merge 8 wave-states per row, form row loss, deterministic per-WG partial.
  if (tid < 16) {
    float mm = -INFINITY, SS = 0.f, AA = 0.f, QQ = 0.f, MM = 0.f;
    #pragma unroll
    for (int w = 0; w < 8; ++w) {
      float m2 = red[w][tid][0];
      float nm = fmaxf(mm, m2);
      SS = SS * __expf(mm - nm) + red[w][tid][1] * __expf(m2 - nm);
      mm = nm;
      AA += red[w][tid][2];
      QQ += red[w][tid][3];
      MM += red[w][tid][4];
    }
    float logS   = logf(SS + EPSV);                    // reproduces log(eps) underflow path
    float meanLP = (AA - (mm + logS) * QQ) / (MM + EPSV);
    float contrib = -(TEMP * meanLP) * (1.0f / (float)NROWS);
    #pragma unroll
    for (int off = 1; off < 16; off <<= 1) contrib += __shfl_xor(contrib, off, 32);
    if (tid == 0) partial[blockIdx.x] = contrib;
  }
}

__global__ void reduce_partials(const float* __restrict__ partial, float* __restrict__ out) {
  __shared__ float buf[256];
  int t = threadIdx.x;
  buf[t] = partial[t];
  __syncthreads();
  for (int s2 = 128; s2 > 0; s2 >>= 1) {
    if (t < s2) buf[t] += buf[t + s2];
    __syncthreads();
  }
  if (t == 0) out[0] = buf[0];
}

extern "C" void kernel_launch(void* const* d_in, const int* in_sizes, int n_in,
                              void* d_out, int out_size, void* d_ws, size_t ws_size,
                              hipStream_t stream) {
  (void)in_sizes; (void)n_in; (void)out_size; (void)ws_size;
  const float* feat   = (const float*)d_in[0];   // [2048, 2, 256] f32
  const int*   labels = (const int*)d_in[1];     // [2048, 80] i32

  uint4* bits    = (uint4*)d_ws;                                   // 32 KB
  float* partial = (float*)((char*)d_ws + B_ROWS * sizeof(uint4)); // 1 KB

  pack_labels<<<B_ROWS / 256, 256, 0, stream>>>(labels, bits);
  supcon_main<<<NROWS / 16, 256, 0, stream>>>(feat, bits, partial);
  reduce_partials<<<1, 256, 0, stream>>>(partial, (float*)d_out);
}